// TransformerBlock_82291573392019
// MI455X (gfx1250) — compile-verified
//
#include <hip/hip_runtime.h>
#include <hip/hip_bf16.h>
#include <math.h>

// ---------------------------------------------------------------------------
// TransformerBlock for MI455X (gfx1250), fp32 end-to-end using the native
// fp32 WMMA op V_WMMA_F32_16X16X4_F32 (exact precision vs the fp32 reference).
//
// Shapes: B=2, N=2048 (8*16*16), C=384, H=8, d=48, hidden=1536.
// All GEMMs are LDS-staged and double-buffered so global latency is paid once
// per 32-deep K-slab instead of once per WMMA.
// ---------------------------------------------------------------------------

typedef __attribute__((ext_vector_type(2))) float v2f;
typedef __attribute__((ext_vector_type(8))) float v8f;

#define WMMA_F32(A, Bf, Cacc) \
    __builtin_amdgcn_wmma_f32_16x16x4_f32(false, (A), false, (Bf), (short)0, (Cacc), false, false)

__device__ __forceinline__ float wave16_max(float v) {
    v = fmaxf(v, __shfl_xor(v, 1));
    v = fmaxf(v, __shfl_xor(v, 2));
    v = fmaxf(v, __shfl_xor(v, 4));
    v = fmaxf(v, __shfl_xor(v, 8));
    return v;
}
__device__ __forceinline__ float wave16_sum(float v) {
    v += __shfl_xor(v, 1);
    v += __shfl_xor(v, 2);
    v += __shfl_xor(v, 4);
    v += __shfl_xor(v, 8);
    return v;
}
__device__ __forceinline__ float wave32_sum(float v) {
    v += __shfl_xor(v, 1);
    v += __shfl_xor(v, 2);
    v += __shfl_xor(v, 4);
    v += __shfl_xor(v, 8);
    v += __shfl_xor(v, 16);
    return v;
}

// ---------------------------------------------------------------------------
// x[B][C][N] -> h[(b*N+n)][C]
// ---------------------------------------------------------------------------
__global__ void tb_transpose_in(const float* __restrict__ x, float* __restrict__ h) {
    const size_t total = (size_t)2 * 384 * 2048;
    for (size_t i = (size_t)blockIdx.x * blockDim.x + threadIdx.x; i < total;
         i += (size_t)gridDim.x * blockDim.x) {
        int c = (int)(i % 384);
        size_t bn = i / 384;
        int n = (int)(bn % 2048);
        int b = (int)(bn / 2048);
        h[i] = x[((size_t)b * 384 + c) * 2048 + n];
    }
}

// h[(b*N+n)][C] -> out[B][C][N]
__global__ void tb_transpose_out(const float* __restrict__ h, float* __restrict__ o) {
    const size_t total = (size_t)2 * 384 * 2048;
    for (size_t i = (size_t)blockIdx.x * blockDim.x + threadIdx.x; i < total;
         i += (size_t)gridDim.x * blockDim.x) {
        int n = (int)(i % 2048);
        size_t bc = i / 2048;
        int c = (int)(bc % 384);
        int b = (int)(bc / 384);
        o[i] = h[((size_t)b * 2048 + n) * 384 + c];
    }
}

// ---------------------------------------------------------------------------
// LayerNorm over C=384: one wave per row, 12 elements/lane, shfl reductions.
// ---------------------------------------------------------------------------
__global__ __launch_bounds__(32) void tb_layernorm(const float* __restrict__ in,
                                                   float* __restrict__ out,
                                                   const float* __restrict__ g,
                                                   const float* __restrict__ b) {
    const int row = blockIdx.x;
    const int lane = threadIdx.x;
    const float* x = in + (size_t)row * 384;
    float xv[12];
    float s = 0.f;
#pragma unroll
    for (int i = 0; i < 12; ++i) { xv[i] = x[lane + 32 * i]; s += xv[i]; }
    s = wave32_sum(s);
    const float mu = s * (1.f / 384.f);
    float vs = 0.f;
#pragma unroll
    for (int i = 0; i < 12; ++i) { float d = xv[i] - mu; vs += d * d; }
    vs = wave32_sum(vs);
    const float rstd = rsqrtf(vs * (1.f / 384.f) + 1e-6f);
    float* o = out + (size_t)row * 384;
#pragma unroll
    for (int i = 0; i < 12; ++i) {
        const int c = lane + 32 * i;
        o[c] = (xv[i] - mu) * rstd * g[c] + b[c];
    }
}

// ---------------------------------------------------------------------------
// fp32 WMMA GEMM: C[M,N] = A[M,K] @ B[K,N]  (+bias, +GELU(exact), +residual)
//
// Block = 128 threads (4 waves), tile = 64(M) x 64(N), K-slab = 32.
// B slab is staged in LDS *transposed* (Bs[n][k], stride 34 so the b64
// fragment reads are bank-conflict-free); A fragments are contiguous b64
// global loads. Both are prefetched one slab ahead (registers), so the
// 32 WMMAs per slab run against LDS while the next slab's global loads fly.
//
// fp32 WMMA fragment layout (ISA 7.12.2):
//   A (16x4):  lane L: m=L%16, holds A[m][kb + 2*(L/16) + {0,1}]
//   B (4x16):  lane L: n=L%16, holds B[kb + 2*(L/16) + {0,1}][n]
//   C/D(16x16):lane L: n=L%16, vgpr r holds D[r + 8*(L/16)][n]
// ---------------------------------------------------------------------------
#define BSTRIDE 34

__global__ __launch_bounds__(128) void tb_wmma_gemm(const float* __restrict__ A,
                                                    const float* __restrict__ Bm,
                                                    float* __restrict__ C,
                                                    const float* __restrict__ bias,
                                                    const float* __restrict__ residual,
                                                    int M, int N, int K, int do_gelu) {
    __shared__ float Bs[2][64 * BSTRIDE];

    const int tid   = threadIdx.x;
    const int lane  = tid & 31;
    const int wave  = tid >> 5;
    const int ln16  = lane & 15;
    const int khalf = (lane >> 4) << 1;  // 0 or 2
    const int m0 = blockIdx.y * 64 + wave * 16;
    const int n0 = blockIdx.x * 64;
    const int nSlab = K >> 5;

    const float* arow = A + (size_t)(m0 + ln16) * K;
    const int bn4 = (tid & 15) << 2;  // 0..60, step 4
    const int bk  = tid >> 4;         // 0..7

    // cooperative global B slab -> registers (coalesced float4 rows)
    auto ldB = [&](int s, float4 br[4]) {
#pragma unroll
        for (int i = 0; i < 4; ++i) {
            const int k = bk + (i << 3);
            br[i] = *(const float4*)&Bm[(size_t)((s << 5) + k) * N + n0 + bn4];
        }
    };
    // registers -> LDS (transposed, padded)
    auto stB = [&](int buf, const float4 br[4]) {
        float* dst = &Bs[buf][0];
#pragma unroll
        for (int i = 0; i < 4; ++i) {
            const int k = bk + (i << 3);
            dst[(bn4 + 0) * BSTRIDE + k] = br[i].x;
            dst[(bn4 + 1) * BSTRIDE + k] = br[i].y;
            dst[(bn4 + 2) * BSTRIDE + k] = br[i].z;
            dst[(bn4 + 3) * BSTRIDE + k] = br[i].w;
        }
    };
    // A fragments for one slab: 8 contiguous b64 loads per lane
    auto ldA = [&](int s, v2f af[8]) {
        const int kb = s << 5;
#pragma unroll
        for (int st = 0; st < 8; ++st)
            af[st] = *(const v2f*)&arow[kb + (st << 2) + khalf];
    };

    v8f acc0 = {}, acc1 = {}, acc2 = {}, acc3 = {};
    float4 breg[4], bregN[4];
    v2f afr[8], afrN[8];

    ldB(0, breg);
    ldA(0, afr);
    stB(0, breg);
    __syncthreads();

    for (int s = 0; s < nSlab; ++s) {
        const int cur = s & 1;
        const bool more = (s + 1) < nSlab;
        if (more) {
            ldB(s + 1, bregN);
            ldA(s + 1, afrN);
            if (s + 2 < nSlab)  // stream hint for the slab after next
                __builtin_prefetch(&Bm[(size_t)((s + 2) << 5) * N + n0 + bn4], 0, 1);
        }
        const float* bs = &Bs[cur][0];
#pragma unroll
        for (int st = 0; st < 8; ++st) {
            const int kk = (st << 2) + khalf;
            v2f b0 = *(const v2f*)&bs[(ln16)      * BSTRIDE + kk];
            v2f b1 = *(const v2f*)&bs[(16 + ln16) * BSTRIDE + kk];
            v2f b2 = *(const v2f*)&bs[(32 + ln16) * BSTRIDE + kk];
            v2f b3 = *(const v2f*)&bs[(48 + ln16) * BSTRIDE + kk];
            acc0 = WMMA_F32(afr[st], b0, acc0);
            acc1 = WMMA_F32(afr[st], b1, acc1);
            acc2 = WMMA_F32(afr[st], b2, acc2);
            acc3 = WMMA_F32(afr[st], b3, acc3);
        }
        if (more) {
            __syncthreads();
            stB(1 - cur, bregN);
#pragma unroll
            for (int st = 0; st < 8; ++st) afr[st] = afrN[st];
            __syncthreads();
        }
    }

    const int mbase = m0 + ((lane >> 4) << 3);
#pragma unroll
    for (int r = 0; r < 8; ++r) {
        const int m = mbase + r;
        float vals[4] = {acc0[r], acc1[r], acc2[r], acc3[r]};
#pragma unroll
        for (int t = 0; t < 4; ++t) {
            const int n = n0 + t * 16 + ln16;
            float v = vals[t];
            if (bias) v += bias[n];
            if (do_gelu) v = 0.5f * v * (1.f + erff(v * 0.70710678118654752f));
            if (residual) v += residual[(size_t)m * N + n];
            C[(size_t)m * N + n] = v;
        }
    }
}

// ---------------------------------------------------------------------------
// Flash attention, fp32 WMMA. One wave per (b, head, 16-query block).
// qkv layout: [(b*2048+n)][1152], inner = which*384 + h*48 + dd.
// out: attn[(b*2048+n)][384], head h occupies cols h*48..h*48+47.
//
// Per key block j: 12 K-fragments are prefetched into registers (batched
// b64 loads) before the S WMMAs; the 16x48 V tile is staged in LDS
// pair-interleaved so P@V fragments are conflict-free ds_load_b64.
// ---------------------------------------------------------------------------
__global__ __launch_bounds__(32) void tb_attention(const float* __restrict__ qkv,
                                                   float* __restrict__ out) {
    __shared__ float pl[256];  // 16x16 P tile (C/D layout -> A layout reroute)
    __shared__ float vl[768];  // V tile, pair-interleaved: [(k>>1)*48+col]*2 + (k&1)

    const int lane  = threadIdx.x;
    const int ln16  = lane & 15;
    const int hi8   = (lane >> 4) << 3;  // 0 or 8
    const int khalf = (lane >> 4) << 1;  // 0 or 2

    const int qb = blockIdx.x & 127;
    const int h  = (blockIdx.x >> 7) & 7;
    const int b  = blockIdx.x >> 10;

    const float SCALE = 0.14433756729740643f;  // 48^-0.5
    const float* base = qkv + (size_t)b * 2048 * 1152;
    const float* qptr = base + h * 48;
    const float* kptr = base + 384 + h * 48;
    const float* vptr = base + 768 + h * 48;

    // Preload q fragments for rows qb*16 + (lane%16): 12 K-steps of K=4.
    v2f qa[12];
    {
        const float* qrow = qptr + (size_t)(qb * 16 + ln16) * 1152;
#pragma unroll
        for (int s = 0; s < 12; ++s)
            qa[s] = *(const v2f*)&qrow[4 * s + khalf];
    }

    v8f acc0 = {}, acc1 = {}, acc2 = {};
    float mrow[8], lrow[8];
#pragma unroll
    for (int r = 0; r < 8; ++r) { mrow[r] = -INFINITY; lrow[r] = 0.f; }

    for (int j = 0; j < 128; ++j) {
        // ---- prefetch K fragments + V tile into registers (loads batch) ----
        v2f kfr[12];
        const float* krow = kptr + (size_t)(j * 16 + ln16) * 1152;
#pragma unroll
        for (int st = 0; st < 12; ++st)
            kfr[st] = *(const v2f*)&krow[4 * st + khalf];

        float vt[24];  // 768 floats cooperatively: idx = i*32 + lane
        const float* vbase = vptr + (size_t)(j * 16) * 1152;
#pragma unroll
        for (int i = 0; i < 24; ++i) {
            const int idx = i * 32 + lane;
            const int vk = idx / 48, vc = idx % 48;
            vt[i] = vbase[(size_t)vk * 1152 + vc];
        }

        // ---- S = q @ k^T ----
        v8f s = {};
#pragma unroll
        for (int st = 0; st < 12; ++st)
            s = WMMA_F32(qa[st], kfr[st], s);

        // ---- online softmax (row r lives in 16 lanes of element r) ----
        float corr[8];
#pragma unroll
        for (int r = 0; r < 8; ++r) {
            const float sv = s[r] * SCALE;
            const float mnew = fmaxf(mrow[r], wave16_max(sv));
            const float p = __expf(sv - mnew);
            const float rs = wave16_sum(p);
            corr[r] = __expf(mrow[r] - mnew);
            lrow[r] = lrow[r] * corr[r] + rs;
            mrow[r] = mnew;
            s[r] = p;
        }
#pragma unroll
        for (int r = 0; r < 8; ++r) {
            acc0[r] *= corr[r];
            acc1[r] *= corr[r];
            acc2[r] *= corr[r];
        }

        // ---- stage P and V in LDS ----
        __syncthreads();
#pragma unroll
        for (int r = 0; r < 8; ++r) pl[(r + hi8) * 16 + ln16] = s[r];
#pragma unroll
        for (int i = 0; i < 24; ++i) {
            const int idx = i * 32 + lane;
            const int vk = idx / 48, vc = idx % 48;
            vl[((vk >> 1) * 48 + vc) * 2 + (vk & 1)] = vt[i];
        }
        __syncthreads();

        // ---- acc += P @ V_block (K=16 => 4 steps; d=48 => 3 col tiles) ----
#pragma unroll
        for (int st = 0; st < 4; ++st) {
            const int kp = 2 * st + (lane >> 4);  // pair index (k'=4st+khalf)
            v2f pa = *(const v2f*)&pl[ln16 * 16 + 4 * st + khalf];
            v2f vb0 = *(const v2f*)&vl[(kp * 48 + ln16) * 2];
            v2f vb1 = *(const v2f*)&vl[(kp * 48 + 16 + ln16) * 2];
            v2f vb2 = *(const v2f*)&vl[(kp * 48 + 32 + ln16) * 2];
            acc0 = WMMA_F32(pa, vb0, acc0);
            acc1 = WMMA_F32(pa, vb1, acc1);
            acc2 = WMMA_F32(pa, vb2, acc2);
        }
    }

    // ---- normalize and write [16 x 48] result ----
#pragma unroll
    for (int r = 0; r < 8; ++r) {
        const int m = qb * 16 + r + hi8;
        const float inv = 1.f / lrow[r];
        float* orow = out + ((size_t)(b * 2048 + m)) * 384 + h * 48;
        orow[ln16]      = acc0[r] * inv;
        orow[16 + ln16] = acc1[r] * inv;
        orow[32 + ln16] = acc2[r] * inv;
    }
}

// ---------------------------------------------------------------------------
// Sequence (ws: A/B = 4096x384, C = 4096x1536; 37.75 MB fp32 total):
//   t_in: x -> A(h0)
//   LN1:  A -> B
//   QKV:  B @ Wqkv -> C            [4096x1152, K=384]
//   ATTN: C -> B (overwrites ln1)
//   WO:   B @ Wo + bo + A -> A     [4096x384, K=384]  (h2)
//   LN2:  A -> B
//   MLP1: B @ W1 + b1m, GELU -> C  [4096x1536, K=384]
//   MLP2: C @ W2 + b2m + A -> B    [4096x384, K=1536]
//   t_out: B -> d_out
// ---------------------------------------------------------------------------
extern "C" void kernel_launch(void* const* d_in, const int* in_sizes, int n_in,
                              void* d_out, int out_size, void* d_ws, size_t ws_size,
                              hipStream_t stream) {
    const float* x    = (const float*)d_in[0];
    const float* g1   = (const float*)d_in[1];
    const float* b1   = (const float*)d_in[2];
    const float* Wqkv = (const float*)d_in[3];
    const float* Wo   = (const float*)d_in[4];
    const float* bo   = (const float*)d_in[5];
    const float* g2   = (const float*)d_in[6];
    const float* b2   = (const float*)d_in[7];
    const float* W1   = (const float*)d_in[8];
    const float* b1m  = (const float*)d_in[9];
    const float* W2   = (const float*)d_in[10];
    const float* b2m  = (const float*)d_in[11];
    float* out = (float*)d_out;

    float* ws = (float*)d_ws;
    const size_t ROWS = 4096;  // B*N
    float* bufA = ws;                   // 4096 x 384
    float* bufB = ws + ROWS * 384;      // 4096 x 384
    float* bufC = ws + 2 * ROWS * 384;  // 4096 x 1536

    tb_transpose_in<<<1024, 256, 0, stream>>>(x, bufA);
    tb_layernorm<<<4096, 32, 0, stream>>>(bufA, bufB, g1, b1);
    tb_wmma_gemm<<<dim3(1152 / 64, 4096 / 64), 128, 0, stream>>>(
        bufB, Wqkv, bufC, nullptr, nullptr, 4096, 1152, 384, 0);
    tb_attention<<<2048, 32, 0, stream>>>(bufC, bufB);
    tb_wmma_gemm<<<dim3(384 / 64, 4096 / 64), 128, 0, stream>>>(
        bufB, Wo, bufA, bo, bufA, 4096, 384, 384, 0);
    tb_layernorm<<<4096, 32, 0, stream>>>(bufA, bufB, g2, b2);
    tb_wmma_gemm<<<dim3(1536 / 64, 4096 / 64), 128, 0, stream>>>(
        bufB, W1, bufC, b1m, nullptr, 4096, 1536, 384, 1);
    tb_wmma_gemm<<<dim3(384 / 64, 4096 / 64), 128, 0, stream>>>(
        bufC, W2, bufB, b2m, bufA, 4096, 384, 1536, 0);
    tb_transpose_out<<<1024, 256, 0, stream>>>(bufB, out);
}